// WaveletDecoder_53910429499945
// MI455X (gfx1250) — compile-verified
//
#include <hip/hip_runtime.h>

// CDNA5 (gfx1250) wave32 WMMA types
typedef __attribute__((ext_vector_type(2))) float v2f;
typedef __attribute__((ext_vector_type(8))) float v8f;

#define INV_SQRT8 0.35355339059327373f  // (1/sqrt(2))^3

// A_full = block-diag(M, M), 16x16, where M[o][s] = (-1)^popcount(o&s)/sqrt(8)
// is the 8x8 normalized Hadamard implementing one 2x2x2 inverse-Haar block.
__device__ __forceinline__ float afull(int o, int k) {
    if ((o < 8) != (k < 8)) return 0.0f;      // off-diagonal blocks are zero
    return (__popc(o & k & 7) & 1) ? -INV_SQRT8 : INV_SQRT8;
}

// One inverse-Haar level:
//   coeffs: (B, 8, D, H, W) f32, subband index s = 4*bD + 2*bH + bW
//   approx: (B, D, H, W) f32 replacing subband 0 (nullptr for coarsest level)
//   out:    (B, 2D, 2H, 2W) f32
// Each wave processes 32 coarse voxels with 4x V_WMMA_F32_16X16X4_F32:
//   D(16x16) = A_full(16x16, K-chunked) x B_full(16x16, K-chunked)
//   B rows 0-7  = subbands 0-7 of positions base+0..15  (group A)
//   B rows 8-15 = subbands 0-7 of positions base+16..31 (group B)
//   D rows 0-7  = outputs o=0..7 for group A; rows 8-15 for group B.
__global__ __launch_bounds__(256) void haar_inv_pass(
    const float* __restrict__ coeffs,
    const float* __restrict__ approx,
    float* __restrict__ out,
    int logD, int logH, int logW, int N) {
    const int tid  = blockIdx.x * blockDim.x + threadIdx.x;
    const int lane = tid & 31;
    const int base = (tid >> 5) << 5;       // wave-uniform
    if (base >= N) return;                  // whole wave exits together

    const int logHW  = logH + logW;
    const int logDHW = logD + logHW;
    const int DHW    = 1 << logDHW;
    const int Wm     = (1 << logW) - 1;
    const int Hm     = (1 << logH) - 1;

    const bool hi = lane >= 16;
    const int  n  = lane & 15;
    const int  p0 = base + n;               // group-A position for this lane
    const int  p1 = p0 + 16;                // group-B position for this lane
    const int  sb = hi ? 2 : 0;             // first subband this lane supplies

    // ---- gather B operand (coefficients) ----
    const int b0 = p0 >> logDHW, sp0 = p0 & (DHW - 1);
    const int b1 = p1 >> logDHW, sp1 = p1 & (DHW - 1);
    const int c0 = (b0 << (3 + logDHW)) + sp0;   // b*8*DHW + spatial
    const int c1 = (b1 << (3 + logDHW)) + sp1;

    // B chunk c (K=4c..4c+3): v0: low half=row 4c, high half=row 4c+2;
    //                         v1: low half=row 4c+1, high half=row 4c+3.
    v2f B0, B1, B2, B3;
    B0.x = coeffs[c0 + (sb + 0) * DHW];
    B0.y = coeffs[c0 + (sb + 1) * DHW];
    B1.x = coeffs[c0 + (sb + 4) * DHW];
    B1.y = coeffs[c0 + (sb + 5) * DHW];
    B2.x = coeffs[c1 + (sb + 0) * DHW];
    B2.y = coeffs[c1 + (sb + 1) * DHW];
    B3.x = coeffs[c1 + (sb + 4) * DHW];
    B3.y = coeffs[c1 + (sb + 5) * DHW];
    if (approx) {                           // overwrite subband 0 (low lanes only)
        const float r0 = approx[p0];
        const float r1 = approx[p1];
        if (!hi) { B0.x = r0; B2.x = r1; }  // lowers to v_cndmask selects
    }

    // ---- build A operand (block-diagonal Hadamard), per ISA 16x4 f32 layout:
    // lanes 0-15: M=lane, v0=K(4c), v1=K(4c+1); lanes 16-31: M=lane-16, K(4c+2/3).
    const int kh  = hi ? 2 : 0;
    v2f A0, A1, A2, A3;
    A0.x = afull(n, 0 + kh);   A0.y = afull(n, 1 + kh);
    A1.x = afull(n, 4 + kh);   A1.y = afull(n, 5 + kh);
    A2.x = afull(n, 8 + kh);   A2.y = afull(n, 9 + kh);
    A3.x = afull(n, 12 + kh);  A3.y = afull(n, 13 + kh);

    // ---- 4x accumulating WMMA: full K=16 ----
    v8f acc = {0.f, 0.f, 0.f, 0.f, 0.f, 0.f, 0.f, 0.f};
    acc = __builtin_amdgcn_wmma_f32_16x16x4_f32(false, A0, false, B0, (short)0, acc, false, false);
    acc = __builtin_amdgcn_wmma_f32_16x16x4_f32(false, A1, false, B1, (short)0, acc, false, false);
    acc = __builtin_amdgcn_wmma_f32_16x16x4_f32(false, A2, false, B2, (short)0, acc, false, false);
    acc = __builtin_amdgcn_wmma_f32_16x16x4_f32(false, A3, false, B3, (short)0, acc, false, false);

    // ---- scatter: lane owns position q = base + lane; acc[r] = output offset r ----
    const int q  = base + lane;
    const int bq = q >> logDHW, spq = q & (DHW - 1);
    const int xq = spq & Wm;
    const int yq = (spq >> logW) & Hm;
    const int zq = spq >> logHW;
    // out (B, 2D, 2H, 2W): z-stride=4HW, y-stride=2W
    const int ystr = 1 << (1 + logW);
    const int zstr = 1 << (2 + logHW);
    const int ob = (bq << (3 + logDHW)) + zq * zstr + yq * ystr + (xq << 1);

    // o = 4i+2j+k; pair (k=0,k=1) -> contiguous float2, fully coalesced stores
    *reinterpret_cast<float2*>(out + ob)               = make_float2(acc[0], acc[1]);
    *reinterpret_cast<float2*>(out + ob + ystr)        = make_float2(acc[2], acc[3]);
    *reinterpret_cast<float2*>(out + ob + zstr)        = make_float2(acc[4], acc[5]);
    *reinterpret_cast<float2*>(out + ob + zstr + ystr) = make_float2(acc[6], acc[7]);
}

extern "C" void kernel_launch(void* const* d_in, const int* in_sizes, int n_in,
                              void* d_out, int out_size, void* d_ws, size_t ws_size,
                              hipStream_t stream) {
    const float* scale0 = (const float*)d_in[0];   // (16, 8, 64, 64, 64)
    const float* scale1 = (const float*)d_in[1];   // (16, 8, 32, 32, 32)
    const float* scale2 = (const float*)d_in[2];   // (16, 8, 16, 16, 16)
    float* out = (float*)d_out;                    // (16, 1, 128, 128, 128)

    const int B = 16;
    float* approx1 = (float*)d_ws;                 // (16, 32, 32, 32)  2 MiB
    float* approx2 = approx1 + (size_t)B * 32 * 32 * 32;  // (16, 64, 64, 64) 16 MiB

    // level 2 (coarsest): (B,8,16^3) -> (B,32^3)
    {
        const int N = B * 16 * 16 * 16;
        haar_inv_pass<<<N / 256, 256, 0, stream>>>(scale2, nullptr, approx1, 4, 4, 4, N);
    }
    // level 1: (B,8,32^3) + approx1 -> (B,64^3)
    {
        const int N = B * 32 * 32 * 32;
        haar_inv_pass<<<N / 256, 256, 0, stream>>>(scale1, approx1, approx2, 5, 5, 5, N);
    }
    // level 0 (finest): (B,8,64^3) + approx2 -> (B,128^3) = d_out
    {
        const int N = B * 64 * 64 * 64;
        haar_inv_pass<<<N / 256, 256, 0, stream>>>(scale0, approx2, out, 6, 6, 6, N);
    }
}